// HiddenMarkovModel_25374666784917
// MI455X (gfx1250) — compile-verified
//
#include <hip/hip_runtime.h>
#include <hip/hip_bf16.h>
#include <math.h>

typedef __attribute__((ext_vector_type(2))) float v2f;
typedef __attribute__((ext_vector_type(8))) float v8f;

#define NSTATES 128
#define DDIM    128
#define BSUB    16
#define TLEN    4096
#define LOG2PI_F 1.8378770664093453f
#define LOG2E_F  1.4426950408889634f
#define LN2_F    0.6931471805599453f

#if __has_builtin(__builtin_amdgcn_exp2f)
#define FAST_EXP2(x) __builtin_amdgcn_exp2f(x)
#else
#define FAST_EXP2(x) exp2f(x)
#endif
#if __has_builtin(__builtin_amdgcn_logf)
#define FAST_LOG2(x) __builtin_amdgcn_logf(x)
#else
#define FAST_LOG2(x) log2f(x)
#endif

// ---------------------------------------------------------------------------
// Kernel 0: parameter preprocessing (one block, 128 threads; negligible cost)
//   G[k][n], k<128: -0.5/var[n][k];  k>=128: mu[n][k-128]/var[n][k-128]
//   CB[n] = -0.5*(sum mu^2/var + sum log var + D*log(2pi))
//   AN[i][j] = softmax(transition, axis=1)   (linear-space transition)
//   LPI[j]   = log_softmax(priors)
// ---------------------------------------------------------------------------
__global__ __launch_bounds__(128)
void hmm_prep_kernel(const float* __restrict__ trans, const float* __restrict__ priors,
                     const float* __restrict__ means, const float* __restrict__ scales,
                     float* __restrict__ G, float* __restrict__ CB,
                     float* __restrict__ AN, float* __restrict__ LPI) {
  const int n = threadIdx.x;           // 0..127 (state index / row index)
  float m2 = 0.f, ld = 0.f;
  for (int d = 0; d < DDIM; ++d) {
    float sr  = scales[n * DDIM + d];
    // stable softplus
    float sp  = fmaxf(sr, 0.f) + log1pf(expf(-fabsf(sr)));
    float var = sp + 1e-6f;
    float inv = 1.f / var;
    float mu  = means[n * DDIM + d];
    G[(size_t)d * NSTATES + n]          = -0.5f * inv;   // coalesced across n
    G[(size_t)(DDIM + d) * NSTATES + n] = mu * inv;
    m2 += mu * mu * inv;
    ld += logf(var);
  }
  CB[n] = -0.5f * (m2 + ld + (float)DDIM * LOG2PI_F);

  // row-softmax of transition (row i == n)
  {
    float mx = -INFINITY;
    for (int j = 0; j < NSTATES; ++j) mx = fmaxf(mx, trans[n * NSTATES + j]);
    float s = 0.f;
    for (int j = 0; j < NSTATES; ++j) s += expf(trans[n * NSTATES + j] - mx);
    float is = 1.f / s;
    for (int j = 0; j < NSTATES; ++j)
      AN[n * NSTATES + j] = expf(trans[n * NSTATES + j] - mx) * is;
  }
  // log_softmax of priors (redundant per-thread compute; tiny)
  {
    float mx = -INFINITY;
    for (int j = 0; j < NSTATES; ++j) mx = fmaxf(mx, priors[j]);
    float s = 0.f;
    for (int j = 0; j < NSTATES; ++j) s += expf(priors[j] - mx);
    LPI[n] = priors[n] - mx - logf(s);
  }
}

// ---------------------------------------------------------------------------
// Kernel 1: emission log-probs via V_WMMA_F32_16X16X4_F32.
//   emis[bt][n] = sum_k A[bt][k] * G[k][n] + CB[n],
//   A[bt][k] = X[bt][k]^2 (k<128) | X[bt][k-128] (k>=128)
// One wave owns a 16-row stripe, all 128 columns (8 N-tiles, 64 K-steps each).
// f32 WMMA operand layout (ISA 7.12.2):
//   A 16x4: lanes 0-15 -> M=0..15, lanes 16-31 -> same M; VGPR0 = K {0 | 2},
//           VGPR1 = K {1 | 3}.  B 4x16 mirrored (N in lanes, K across VGPR/half).
//   C/D: VGPR r: lanes 0-15 -> M=r, N=lane; lanes 16-31 -> M=r+8, N=lane-16.
// ---------------------------------------------------------------------------
__global__ __launch_bounds__(128)
void hmm_emis_kernel(const float* __restrict__ X, const float* __restrict__ G,
                     const float* __restrict__ CB, float* __restrict__ EMIS) {
  const int lane    = threadIdx.x & 31;
  const int wave    = threadIdx.x >> 5;
  const int rowTile = (blockIdx.x * 4 + wave) * 16;
  const int nlo     = lane & 15;
  const int khalf   = lane >> 4;
  const float* Xrow = X + (size_t)(rowTile + nlo) * DDIM;   // this lane's A row

  for (int j = 0; j < 8; ++j) {                             // N tile
    v8f acc = {0.f, 0.f, 0.f, 0.f, 0.f, 0.f, 0.f, 0.f};
    const float* Gj = G + j * 16 + nlo;
#pragma unroll 8
    for (int ks = 0; ks < 32; ++ks) {                       // k in [0,128): x^2 terms
      int k0 = ks * 4 + khalf * 2;
      float x0 = Xrow[k0], x1 = Xrow[k0 + 1];
      v2f a;  a.x = x0 * x0;                a.y = x1 * x1;
      v2f bb; bb.x = Gj[(size_t)k0 * NSTATES]; bb.y = Gj[(size_t)(k0 + 1) * NSTATES];
      acc = __builtin_amdgcn_wmma_f32_16x16x4_f32(false, a, false, bb,
                                                  (short)0, acc, false, false);
    }
#pragma unroll 8
    for (int ks = 32; ks < 64; ++ks) {                      // k in [128,256): x terms
      int k0 = ks * 4 + khalf * 2;
      v2f a;  a.x = Xrow[k0 - 128];         a.y = Xrow[k0 - 127];
      v2f bb; bb.x = Gj[(size_t)k0 * NSTATES]; bb.y = Gj[(size_t)(k0 + 1) * NSTATES];
      acc = __builtin_amdgcn_wmma_f32_16x16x4_f32(false, a, false, bb,
                                                  (short)0, acc, false, false);
    }
    float cb = CB[j * 16 + nlo];
#pragma unroll
    for (int r = 0; r < 8; ++r) {
      int mm = rowTile + r + khalf * 8;
      EMIS[(size_t)mm * NSTATES + j * 16 + nlo] = acc[r] + cb;
    }
  }
}

// ---------------------------------------------------------------------------
// Kernel 2: serial forward recursion, one workgroup per subject b.
// Thread j holds column AN[:,j] in 128 VGPRs for all 4095 steps.
//   m = max_i alpha_i;  v_i = exp(alpha_i - m)   (shared via LDS)
//   alpha'_j = e[t,j] + m + log( sum_i v_i * AN[i,j] )
// Exactly 2 barriers per step; single-buffer LDS is race-free given the
// barrier ordering (all readers pass a barrier before the next writer writes).
// ---------------------------------------------------------------------------
__global__ __launch_bounds__(128)
void hmm_fwd_kernel(const float* __restrict__ AN, const float* __restrict__ LPI,
                    const float* __restrict__ EMIS, float* __restrict__ RES) {
  __shared__ __align__(16) float vbuf[NSTATES];
  __shared__ float wmax[4];
  const int j = threadIdx.x, b = blockIdx.x;
  const int wid = j >> 5, lane = j & 31;

  float a[NSTATES];
#pragma unroll
  for (int i = 0; i < NSTATES; ++i) a[i] = AN[i * NSTATES + j];   // coalesced

  const float* eb = EMIS + (size_t)b * TLEN * NSTATES;
  float alpha = LPI[j] + eb[j];

  for (int t = 1; t < TLEN; ++t) {
    float e = eb[(size_t)t * NSTATES + j];                 // issue early (L2 hit)
    __builtin_prefetch(&eb[(size_t)(t + 1) * NSTATES + j], 0, 0); // speculative
    // wave-level max (wave32), then 4-wave combine through LDS
    float m = alpha;
    m = fmaxf(m, __shfl_xor(m, 16, 32));
    m = fmaxf(m, __shfl_xor(m, 8, 32));
    m = fmaxf(m, __shfl_xor(m, 4, 32));
    m = fmaxf(m, __shfl_xor(m, 2, 32));
    m = fmaxf(m, __shfl_xor(m, 1, 32));
    if (lane == 0) wmax[wid] = m;
    __syncthreads();
    m = fmaxf(fmaxf(wmax[0], wmax[1]), fmaxf(wmax[2], wmax[3]));
    vbuf[j] = FAST_EXP2((alpha - m) * LOG2E_F);
    __syncthreads();
    // dot(column j of AN, v) — 4 accumulators break the FMA dependency chain
    float s0 = 0.f, s1 = 0.f, s2 = 0.f, s3 = 0.f;
#pragma unroll
    for (int q = 0; q < NSTATES / 4; ++q) {
      float4 vv = ((const float4*)vbuf)[q];                // ds_load_b128 broadcast
      s0 = fmaf(vv.x, a[4 * q + 0], s0);
      s1 = fmaf(vv.y, a[4 * q + 1], s1);
      s2 = fmaf(vv.z, a[4 * q + 2], s2);
      s3 = fmaf(vv.w, a[4 * q + 3], s3);
    }
    alpha = e + m + FAST_LOG2((s0 + s1) + (s2 + s3)) * LN2_F;
  }

  // final logsumexp over states for this subject
  float m = alpha;
  m = fmaxf(m, __shfl_xor(m, 16, 32));
  m = fmaxf(m, __shfl_xor(m, 8, 32));
  m = fmaxf(m, __shfl_xor(m, 4, 32));
  m = fmaxf(m, __shfl_xor(m, 2, 32));
  m = fmaxf(m, __shfl_xor(m, 1, 32));
  if (lane == 0) wmax[wid] = m;
  __syncthreads();
  m = fmaxf(fmaxf(wmax[0], wmax[1]), fmaxf(wmax[2], wmax[3]));
  vbuf[j] = FAST_EXP2((alpha - m) * LOG2E_F);
  __syncthreads();
  if (j == 0) {
    float s = 0.f;
    for (int i = 0; i < NSTATES; ++i) s += vbuf[i];
    RES[b] = m + FAST_LOG2(s) * LN2_F;
  }
}

// Kernel 3: deterministic fixed-order sum over the 16 subjects (no atomics).
__global__ void hmm_finish_kernel(const float* __restrict__ RES, float* __restrict__ out) {
  if (threadIdx.x == 0 && blockIdx.x == 0) {
    float s = 0.f;
#pragma unroll
    for (int b = 0; b < BSUB; ++b) s += RES[b];
    out[0] = s;
  }
}

extern "C" void kernel_launch(void* const* d_in, const int* in_sizes, int n_in,
                              void* d_out, int out_size, void* d_ws, size_t ws_size,
                              hipStream_t stream) {
  const float* X      = (const float*)d_in[0];   // [B,T,D]
  const float* trans  = (const float*)d_in[1];   // [N,N]
  const float* priors = (const float*)d_in[2];   // [N]
  const float* means  = (const float*)d_in[3];   // [N,D]
  const float* scales = (const float*)d_in[4];   // [N,D]
  float* out = (float*)d_out;
  float* ws  = (float*)d_ws;

  // workspace layout (floats)
  float* G    = ws;              // [256,128]  folded GEMM B-matrix
  float* CB   = ws + 32768;      // [128]      per-state bias
  float* AN   = ws + 32896;      // [128,128]  softmax(transition)
  float* LPI  = ws + 49280;      // [128]      log_softmax(priors)
  float* RES  = ws + 49408;      // [16]       per-subject log-likelihood
  float* EMIS = ws + 49536;      // [B*T,128]  emission log-probs (~33.5 MB, L2-resident)

  hmm_prep_kernel<<<1, 128, 0, stream>>>(trans, priors, means, scales, G, CB, AN, LPI);
  hmm_emis_kernel<<<(BSUB * TLEN) / 64, 128, 0, stream>>>(X, G, CB, EMIS);
  hmm_fwd_kernel<<<BSUB, 128, 0, stream>>>(AN, LPI, EMIS, RES);
  hmm_finish_kernel<<<1, 32, 0, stream>>>(RES, out);
}